// minLSTMCell_15547781611890
// MI455X (gfx1250) — compile-verified
//
#include <hip/hip_runtime.h>
#include <math.h>

// Problem constants (from setup_inputs): B=4, S=8192, D=H=1024, G=128 -> 8x8 blocks.
#define EPSV   1e-8f
#define GROUPS 128
#define Bdim   4
#define Sdim   8192
#define Ddim   1024
#define Hdim   1024
#define NCHUNK 64
#define CLEN   (Sdim / NCHUNK)   // 128

typedef float v2f __attribute__((ext_vector_type(2)));
typedef float v8f __attribute__((ext_vector_type(8)));

__device__ __forceinline__ float softplus_f(float x) {
  // stable softplus: max(x,0) + log1p(exp(-|x|))
  return fmaxf(x, 0.f) + log1pf(expf(-fabsf(x)));
}

__device__ __forceinline__ float logaddexp_f(float a, float b) {
  float mx = fmaxf(a, b), mn = fminf(a, b);
  if (mx == -INFINITY) return -INFINITY;
  return mx + log1pf(expf(mn - mx));
}

__device__ __forceinline__ float log_g_f(float x) {
  return (x >= 0.f) ? logf(x + 0.5f + EPSV) : (-softplus_f(-x) + EPSV);
}

// Load one v2f B-operand slice of an 8x8 group weight, zeroing lanes outside
// this group's N-range (block-diagonal packing of 2 groups into one 16x16 tile).
__device__ __forceinline__ v2f ldw(const float* __restrict__ Wg, int cn, int kc,
                                   int hi, bool valid) {
  v2f w = *(const v2f*)(Wg + cn * 8 + kc * 4 + 2 * hi);
  v2f r;
  r.x = valid ? w.x : 0.f;
  r.y = valid ? w.y : 0.f;
  return r;
}

// 16 tokens x 16 outputs (= groups g0,g0+1) of one gate via 4 chained
// V_WMMA_F32_16X16X4_F32 (full fp32 accuracy; K=8 per group done as 2xK=4).
__device__ __forceinline__ v8f gate16(v2f a00, v2f a01, v2f a10, v2f a11,
                                      const float* __restrict__ W, int g0,
                                      int n, int hi) {
  const int  cn = n & 7;
  const bool lo = (n < 8);
  const float* Wg0 = W + (size_t)g0 * 64;
  const float* Wg1 = Wg0 + 64;
  v2f b00 = ldw(Wg0, cn, 0, hi, lo);
  v2f b01 = ldw(Wg0, cn, 1, hi, lo);
  v2f b10 = ldw(Wg1, cn, 0, hi, !lo);
  v2f b11 = ldw(Wg1, cn, 1, hi, !lo);
  v8f acc = {};
  acc = __builtin_amdgcn_wmma_f32_16x16x4_f32(false, a00, false, b00, (short)0, acc, false, false);
  acc = __builtin_amdgcn_wmma_f32_16x16x4_f32(false, a01, false, b01, (short)0, acc, false, false);
  acc = __builtin_amdgcn_wmma_f32_16x16x4_f32(false, a10, false, b10, (short)0, acc, false, false);
  acc = __builtin_amdgcn_wmma_f32_16x16x4_f32(false, a11, false, b11, (short)0, acc, false, false);
  return acc;
}

// Kernel A: grouped gemms (WMMA f32) + log-space gate math.
// One wave32 handles 16 tokens x 2 groups (16 output channels) for all 3 gates.
__global__ void __launch_bounds__(256) gates_wmma(
    const float* __restrict__ x, const float* __restrict__ Wf,
    const float* __restrict__ Wi, const float* __restrict__ Wh,
    float* __restrict__ lf_out, float* __restrict__ lv_out) {
  const int lane = threadIdx.x & 31;
  const int wv   = threadIdx.x >> 5;
  const int gid  = blockIdx.x * 8 + wv;        // global wave id
  const int NPAIR = GROUPS / 2;                // 64 group-pairs
  const int gp = gid % NPAIR;
  const int tt = gid / NPAIR;                  // 16-token tile id
  const int s0 = (tt * 16) % Sdim;
  const int b  = (tt * 16) / Sdim;

  const int n  = lane & 15;                    // N (output col) / A row M
  const int hi = lane >> 4;                    // half-wave select (K pairs / M+8)

  // A-matrix 16x4 fp32 layout: lanes 0-15 hold K={0,1}, lanes 16-31 K={2,3}.
  const float* xrow =
      x + ((size_t)b * Sdim + (size_t)(s0 + n)) * (size_t)Ddim + (size_t)gp * 16;
  v2f a00 = *(const v2f*)(xrow + 0  + 2 * hi);   // g0, k 0..3
  v2f a01 = *(const v2f*)(xrow + 4  + 2 * hi);   // g0, k 4..7
  v2f a10 = *(const v2f*)(xrow + 8  + 2 * hi);   // g1, k 0..3
  v2f a11 = *(const v2f*)(xrow + 12 + 2 * hi);   // g1, k 4..7

  const int g0 = gp * 2;
  v8f fg = gate16(a00, a01, a10, a11, Wf, g0, n, hi);
  v8f ig = gate16(a00, a01, a10, a11, Wi, g0, n, hi);
  v8f hg = gate16(a00, a01, a10, a11, Wh, g0, n, hi);

  // D layout: VGPR r -> token s0 + r + 8*hi, channel gp*16 + n.
  const size_t baseoff =
      ((size_t)b * Sdim + (size_t)(s0 + 8 * hi)) * (size_t)Hdim + (size_t)gp * 16 + n;
#pragma unroll
  for (int r = 0; r < 8; ++r) {
    float f  = fg[r];
    float i  = ig[r];
    float hd = hg[r];
    float diff  = softplus_f(-f) - softplus_f(-i);
    float logf_ = -softplus_f(diff);
    float logi_ = -softplus_f(-diff);
    float lv    = logi_ + log_g_f(hd);
    size_t off = baseoff + (size_t)r * Hdim;
    lf_out[off] = logf_;
    lv_out[off] = lv;
  }
}

// Kernel B: per-(channel, chunk) local log-scan -> chunk summaries.
__global__ void __launch_bounds__(256) chunk_local(
    const float* __restrict__ lf, const float* __restrict__ lv,
    float* __restrict__ chA, float* __restrict__ chM) {
  int idx = blockIdx.x * 256 + threadIdx.x;    // [0, B*NCHUNK*H)
  int h  = idx % Hdim;
  int t2 = idx / Hdim;
  int c  = t2 % NCHUNK;
  int b  = t2 / NCHUNK;
  size_t base = ((size_t)b * Sdim + (size_t)c * CLEN) * (size_t)Hdim + h;
  float a = 0.f, l = -INFINITY;
  for (int t = 0; t < CLEN; ++t) {
    size_t off = base + (size_t)t * Hdim;
    __builtin_prefetch(lf + off + (size_t)8 * Hdim, 0, 0);
    __builtin_prefetch(lv + off + (size_t)8 * Hdim, 0, 0);
    float lfv = lf[off];
    float lvv = lv[off];
    a += lfv + EPSV;
    l = logaddexp_f(l, lvv - a);
  }
  int sidx = c * (Bdim * Hdim) + b * Hdim + h; // chunk-major for carry kernel
  chA[sidx] = a;
  chM[sidx] = l;
}

// Kernel B2: tiny sequential carry scan over the 64 chunks per channel.
// Emits Q_c = G_c + P_c where P_c = sum_{j<c} A_j, G_c = logsumexp_{j<c}(m_j - P_j).
__global__ void __launch_bounds__(256) carry_scan(
    const float* __restrict__ chA, const float* __restrict__ chM,
    float* __restrict__ cQ) {
  int chan = blockIdx.x * 256 + threadIdx.x;
  const int nchan = Bdim * Hdim;
  if (chan >= nchan) return;
  float P = 0.f, G = -INFINITY;
  for (int c = 0; c < NCHUNK; ++c) {
    cQ[c * nchan + chan] = G + P;                       // -inf for c==0
    G = logaddexp_f(G, chM[c * nchan + chan] - P);
    P += chA[c * nchan + chan];
  }
}

// Kernel C: re-walk each chunk, fold carry: h_t = exp(a_loc + logaddexp(Q, lse_loc)).
__global__ void __launch_bounds__(256) apply_scan(
    const float* __restrict__ lf, const float* __restrict__ lv,
    const float* __restrict__ cQ, float* __restrict__ out) {
  int idx = blockIdx.x * 256 + threadIdx.x;
  int h  = idx % Hdim;
  int t2 = idx / Hdim;
  int c  = t2 % NCHUNK;
  int b  = t2 / NCHUNK;
  size_t base = ((size_t)b * Sdim + (size_t)c * CLEN) * (size_t)Hdim + h;
  float a = 0.f;
  float R = cQ[c * (Bdim * Hdim) + b * Hdim + h];
  for (int t = 0; t < CLEN; ++t) {
    size_t off = base + (size_t)t * Hdim;
    __builtin_prefetch(lf + off + (size_t)8 * Hdim, 0, 0);
    __builtin_prefetch(lv + off + (size_t)8 * Hdim, 0, 0);
    float lfv = lf[off];
    float lvv = lv[off];
    a += lfv + EPSV;
    R = logaddexp_f(R, lvv - a);
    out[off] = expf(a + R);
  }
}

extern "C" void kernel_launch(void* const* d_in, const int* in_sizes, int n_in,
                              void* d_out, int out_size, void* d_ws, size_t ws_size,
                              hipStream_t stream) {
  (void)in_sizes; (void)n_in; (void)out_size; (void)ws_size;
  const float* x  = (const float*)d_in[0];
  const float* Wf = (const float*)d_in[1];
  const float* Wi = (const float*)d_in[2];
  const float* Wh = (const float*)d_in[3];
  float* out = (float*)d_out;

  const size_t NBSH = (size_t)Bdim * Sdim * Hdim;          // 33.5M floats
  const size_t NSUM = (size_t)NCHUNK * Bdim * Hdim;        // 262144 floats
  float* lf  = (float*)d_ws;
  float* lv  = lf + NBSH;
  float* chA = lv + NBSH;
  float* chM = chA + NSUM;
  float* cQ  = chM + NSUM;

  const int gate_blocks = (Bdim * Sdim / 16) * (GROUPS / 2) / 8;  // 16384
  gates_wmma<<<gate_blocks, 256, 0, stream>>>(x, Wf, Wi, Wh, lf, lv);

  const int scan_threads = Bdim * NCHUNK * Hdim;                   // 262144
  chunk_local<<<scan_threads / 256, 256, 0, stream>>>(lf, lv, chA, chM);
  carry_scan<<<(Bdim * Hdim + 255) / 256, 256, 0, stream>>>(chA, chM, cQ);
  apply_scan<<<scan_threads / 256, 256, 0, stream>>>(lf, lv, cQ, out);
}